// ArvcLoss_43946105372691
// MI455X (gfx1250) — compile-verified
//
#include <hip/hip_runtime.h>
#include <stdint.h>

// ---------------------------------------------------------------------------
// Problem constants (reference: B=8, N=2,000,000, C=16)
// ---------------------------------------------------------------------------
#define BATCHES 8
#define NCLASS 16
#define HLOG 22
#define HSIZE (1u << HLOG)          // 4,194,304 slots per batch (load < 0.5)
#define HMASK (HSIZE - 1u)

typedef float v2f __attribute__((ext_vector_type(2)));
typedef float v8f __attribute__((ext_vector_type(8)));

// Order-preserving mapping float -> uint32 (ascending), with -0 -> +0 first.
__device__ __forceinline__ uint32_t sortable_bits(float f) {
    uint32_t b = __float_as_uint(f);
    if (b == 0x80000000u) b = 0u;                       // canonicalize -0.0
    return (b & 0x80000000u) ? ~b : (b | 0x80000000u);
}

__device__ __forceinline__ uint64_t mix64(uint64_t k) {
    k *= 0x9E3779B97F4A7C15ull;
    k ^= k >> 29;
    k *= 0xBF58476D1CE4E5B9ull;
    k ^= k >> 32;
    return k;
}

// ---------------------------------------------------------------------------
// Kernel 1: per-(batch,class) sum & count via V_WMMA_F32_16X16X4_F32.
// Per WMMA (4 elements e..e+3):
//   A (16x4): row0 = values, row1 = ones, rows 2-15 = 0
//   B (4x16): B[k][c] = (t[e+k] == c)
//   D row0 accumulates per-class sums, row1 per-class counts (in C operand).
//
// VGPR layouts (wave32):
//   A: lane m (m<16) holds row m K=0,1 ; lane m+16 holds row m K=2,3
//   B: lanes 0-15 hold rows K=0(.x),K=1(.y); lanes 16-31 rows K=2(.x),K=3(.y)
//   D: VGPR r, lanes 0-15 = (M=r, N=lane)
//
// Branchless: each lane fetches exactly the two elements its half needs via a
// per-lane ds_bpermute index (i0 = 4s + (lane>=16 ? 2 : 0)).
// ---------------------------------------------------------------------------
__global__ void sumcount_wmma_kernel(const float* __restrict__ x,
                                     const int* __restrict__ t,
                                     float* __restrict__ ssum,
                                     float* __restrict__ scnt,
                                     int n_per_batch, int chunks) {
    const int b = blockIdx.y;
    const float* xb = x + (size_t)b * n_per_batch;
    const int*   tb = t + (size_t)b * n_per_batch;

    const int lane   = threadIdx.x & 31;
    const int wave   = (blockIdx.x * blockDim.x + threadIdx.x) >> 5;
    const int nwaves = (gridDim.x * blockDim.x) >> 5;

    v8f acc = {0.f, 0.f, 0.f, 0.f, 0.f, 0.f, 0.f, 0.f};

    const int col = lane & 15;          // class column for B / N index for D
    const int klo = (lane >= 16) ? 2 : 0;

    for (int chunk = wave; chunk < chunks; chunk += nwaves) {
        const int base = chunk * 32;
        const float xv = xb[base + lane];
        const int   tv = tb[base + lane];

#pragma unroll
        for (int s = 0; s < 8; ++s) {
            const int i0 = 4 * s + klo;           // per-lane varying index
            const float xa = __shfl(xv, i0, 32);      // elem i0   (this half)
            const float xc = __shfl(xv, i0 + 1, 32);  // elem i0+1
            const int   ta = __shfl(tv, i0, 32);
            const int   tc = __shfl(tv, i0 + 1, 32);

            // A operand: col==0 -> value row, col==1 -> ones row, else 0.
            const float ax = (col == 0) ? xa : ((col == 1) ? 1.f : 0.f);
            const float ay = (col == 0) ? xc : ((col == 1) ? 1.f : 0.f);
            v2f a = {ax, ay};

            // B operand: one-hot of the two labels vs this lane's class.
            v2f bm = {(ta == col) ? 1.f : 0.f, (tc == col) ? 1.f : 0.f};

            acc = __builtin_amdgcn_wmma_f32_16x16x4_f32(
                false, a, false, bm, (short)0, acc, false, false);
        }
    }

    // D row 0 = sums, row 1 = counts (lanes 0-15, N = lane).
    if (lane < 16) {
        atomicAdd(&ssum[b * NCLASS + lane], acc[0]);
        atomicAdd(&scnt[b * NCLASS + lane], acc[1]);
    }

    // Tail elements (n_per_batch % 32), handled scalar by wave 0.
    const int tail_base = chunks * 32;
    const int rem = n_per_batch - tail_base;
    if (wave == 0 && lane < rem) {
        const int tt = tb[tail_base + lane];
        atomicAdd(&ssum[b * NCLASS + tt], xb[tail_base + lane]);
        atomicAdd(&scnt[b * NCLASS + tt], 1.0f);
    }
}

// ---------------------------------------------------------------------------
// Kernel 2: hash-insert (exact multiplicity counting). Table lives in L2.
// key = ((t+1) << 32) | sortable(x); empty slot = 0.
// ---------------------------------------------------------------------------
__global__ void hash_insert_kernel(const float* __restrict__ xb,
                                   const int* __restrict__ tb,
                                   unsigned long long* __restrict__ hkeys,
                                   unsigned int* __restrict__ hcnts, int n) {
    const int stride = gridDim.x * blockDim.x;
    for (int i = blockIdx.x * blockDim.x + threadIdx.x; i < n; i += stride) {
        const uint32_t u = sortable_bits(xb[i]);
        const uint64_t key =
            ((uint64_t)(uint32_t)(tb[i] + 1) << 32) | (uint64_t)u;
        uint32_t h = (uint32_t)(mix64(key) & HMASK);
        for (;;) {
            unsigned long long old =
                atomicCAS(&hkeys[h], 0ull, (unsigned long long)key);
            if (old == 0ull || old == (unsigned long long)key) {
                atomicAdd(&hcnts[h], 1u);
                break;
            }
            h = (h + 1u) & HMASK;
        }
    }
}

// ---------------------------------------------------------------------------
// Kernel 3: scan table -> per-group mode via atomicMax of (cnt<<32 | ~u).
// Max count wins; ties pick max(~u) == min(u) == smallest value.
// ---------------------------------------------------------------------------
__global__ void mode_scan_kernel(const unsigned long long* __restrict__ hkeys,
                                 const unsigned int* __restrict__ hcnts,
                                 unsigned long long* __restrict__ modepack,
                                 int b) {
    const int stride = gridDim.x * blockDim.x;
    for (uint32_t s = blockIdx.x * blockDim.x + threadIdx.x; s < HSIZE;
         s += stride) {
        const unsigned long long key = hkeys[s];
        if (key == 0ull) continue;
        const int lab = (int)(key >> 32) - 1;
        const uint32_t u = (uint32_t)key;
        const unsigned long long pack =
            ((unsigned long long)hcnts[s] << 32) | (unsigned long long)(~u);
        atomicMax(&modepack[b * NCLASS + lab], pack);
    }
}

// ---------------------------------------------------------------------------
// Kernel 4: finalize scalar = sum_g scnt_g * (mean_g - mode_g) / (B*N)
// ---------------------------------------------------------------------------
__global__ void finalize_kernel(const float* __restrict__ ssum,
                                const float* __restrict__ scnt,
                                const unsigned long long* __restrict__ modepack,
                                float* __restrict__ out, float inv_total) {
    __shared__ float red[BATCHES * NCLASS];
    const int g = threadIdx.x;                  // 0..127
    float contrib = 0.f;
    const float sc = scnt[g];
    if (sc > 0.f) {
        const uint32_t nu = (uint32_t)(modepack[g] & 0xFFFFFFFFull);
        const uint32_t u = ~nu;
        const uint32_t bits =
            (u & 0x80000000u) ? (u ^ 0x80000000u) : ~u;   // invert sortable map
        const float mode = __uint_as_float(bits);
        contrib = sc * (ssum[g] / sc - mode);
    }
    red[g] = contrib;
    __syncthreads();
    for (int off = (BATCHES * NCLASS) / 2; off > 0; off >>= 1) {
        if (g < off) red[g] += red[g + off];
        __syncthreads();
    }
    if (g == 0) out[0] = red[0] * inv_total;
}

// ---------------------------------------------------------------------------
extern "C" void kernel_launch(void* const* d_in, const int* in_sizes, int n_in,
                              void* d_out, int out_size, void* d_ws,
                              size_t ws_size, hipStream_t stream) {
    const float* x = (const float*)d_in[0];
    const int*   t = (const int*)d_in[1];
    const int total = in_sizes[0];
    const int N = total / BATCHES;
    const int chunks = N / 32;

    // Workspace layout
    float* ssum = (float*)d_ws;                                   // 128 f32
    float* scnt = ssum + BATCHES * NCLASS;                        // 128 f32
    unsigned long long* modepack =
        (unsigned long long*)(scnt + BATCHES * NCLASS);           // 128 u64
    unsigned long long* hkeys = modepack + BATCHES * NCLASS;      // HSIZE u64
    unsigned int* hcnts = (unsigned int*)(hkeys + HSIZE);         // HSIZE u32

    // Zero the per-group stats (sums/cnts + modepack).
    hipMemsetAsync(d_ws, 0,
                   BATCHES * NCLASS * (2 * sizeof(float) + sizeof(uint64_t)),
                   stream);

    // Sum/count via WMMA (independent of the hash passes).
    {
        dim3 grid(256, BATCHES);
        sumcount_wmma_kernel<<<grid, 256, 0, stream>>>(x, t, ssum, scnt, N,
                                                       chunks);
    }

    // Per-batch exact duplicate counting + mode extraction (table fits in L2).
    for (int b = 0; b < BATCHES; ++b) {
        hipMemsetAsync(hkeys, 0, (size_t)HSIZE * sizeof(unsigned long long),
                       stream);
        hipMemsetAsync(hcnts, 0, (size_t)HSIZE * sizeof(unsigned int), stream);
        hash_insert_kernel<<<4096, 256, 0, stream>>>(
            x + (size_t)b * N, t + (size_t)b * N, hkeys, hcnts, N);
        mode_scan_kernel<<<4096, 256, 0, stream>>>(hkeys, hcnts, modepack, b);
    }

    finalize_kernel<<<1, BATCHES * NCLASS, 0, stream>>>(
        ssum, scnt, modepack, (float*)d_out, 1.0f / (float)total);
}